// AdaptiveBoundaryRankingLoss_59803124630165
// MI455X (gfx1250) — compile-verified
//
#include <hip/hip_runtime.h>
#include <stdint.h>

#define BETA   0.3f
#define GAMMA  0.1f
#define BN     8192
#define THREADS 256
#define ROWS_PER_BLOCK 128            // 8 waves x 16 rows each
#define NBLOCKS (BN / ROWS_PER_BLOCK) // 64
#define CHUNK   2048                  // columns staged in LDS per iteration (2 x 8KB)

typedef __attribute__((ext_vector_type(2))) float v2f;
typedef __attribute__((ext_vector_type(8))) float v8f;

// Pair kernel: each wave owns a 16-row strip; for every 16-column tile it builds
//   pred_diff[m,n] = pred[i0+m] - pred[j0+n]
//   targ_diff[m,n] = target[i0+m] - target[j0+n]
// via V_WMMA_F32_16X16X4_F32 as a rank-1(+C) product:
//   A[m,k] = x[m] for ALL k  (lanes 0-15 / 16-31 both hold x[lane&15])
//   B[k,n] = 0.25 everywhere -> sum_k A*B = x[m] (up to 1 ulp), independent
//                               of B's K-slot ordering
//   C[m,n] = -y[n]           -> known C layout: every C VGPR = -y[lane&15]
// Nonlinear boundary + hinge on VALU over the 8 D-layout regs:
//   - divide replaced by hardware V_RCP_F32 (~1 ulp)
//   - sign(t_diff)*pred_diff done as a sign-bit XOR transfer (2 bitwise ops
//     instead of 2 cmps + 2 cndmasks per element)
__global__ __launch_bounds__(THREADS)
void abrl_pairs(const float* __restrict__ pred,
                const float* __restrict__ target,
                float* __restrict__ partials) {
  __shared__ float sp[CHUNK];
  __shared__ float st[CHUNK];
  __shared__ float wred[THREADS / 32];

  const int tid  = threadIdx.x;
  const int wave = tid >> 5;
  const int lane = tid & 31;
  const int ln16 = lane & 15;

  const int i0 = blockIdx.x * ROWS_PER_BLOCK + wave * 16;

  const float px = pred[i0 + ln16];
  const float tx = target[i0 + ln16];

  v2f ap;  ap[0]  = px;    ap[1]  = px;     // A (pred rows), all K slots equal
  v2f at2; at2[0] = tx;    at2[1] = tx;     // A (target rows)
  v2f bq;  bq[0]  = 0.25f; bq[1]  = 0.25f;  // B: column sums == 1

  v8f acc = {0.f,0.f,0.f,0.f,0.f,0.f,0.f,0.f};

  const unsigned lds_p = (unsigned)(uintptr_t)sp;  // low 32 bits of generic
  const unsigned lds_t = (unsigned)(uintptr_t)st;  // LDS pointer = LDS offset

  for (int c0 = 0; c0 < BN; c0 += CHUNK) {
    __syncthreads();  // previous chunk fully consumed before overwrite

    // ---- CDNA5 async global->LDS staging, tracked by ASYNCcnt ----
    #pragma unroll
    for (int pass = 0; pass < (CHUNK / (THREADS * 4)); ++pass) {  // 2 passes
      const int fofs = (pass * THREADS + tid) * 4;                // float units (one float4 per thread)
      const unsigned lofs = (unsigned)(fofs * 4);                 // bytes
      const float* gp = pred   + c0 + fofs;
      const float* gt = target + c0 + fofs;
      asm volatile("global_load_async_to_lds_b128 %0, %1, off"
                   :: "v"(lds_p + lofs), "v"(gp) : "memory");
      asm volatile("global_load_async_to_lds_b128 %0, %1, off"
                   :: "v"(lds_t + lofs), "v"(gt) : "memory");
    }
    asm volatile("s_wait_asynccnt 0" ::: "memory");
    __syncthreads();

    for (int j = 0; j < CHUNK; j += 16) {
      const float py = sp[j + ln16];   // LDS broadcast reads
      const float ty = st[j + ln16];
      v8f cp, ct;
      #pragma unroll
      for (int e = 0; e < 8; ++e) { cp[e] = -py; ct[e] = -ty; }

      v8f pd = __builtin_amdgcn_wmma_f32_16x16x4_f32(
                  false, ap,  false, bq, (short)0, cp, false, false);
      v8f td = __builtin_amdgcn_wmma_f32_16x16x4_f32(
                  false, at2, false, bq, (short)0, ct, false, false);

      #pragma unroll
      for (int e = 0; e < 8; ++e) {
        const float dt  = td[e];
        const float adt = fabsf(dt);
        const float den = fmaf(GAMMA, adt, 1.0f);
        // Weber-Fechner boundary via hardware V_RCP_F32 (~1 ulp)
        const float bnd = (BETA * adt) * __builtin_amdgcn_rcpf(den);
        // pred_diff * sign(t_diff): transfer dt's sign bit onto pd
        const unsigned sb = __float_as_uint(dt) & 0x80000000u;
        const float ps = __uint_as_float(__float_as_uint(pd[e]) ^ sb);
        acc[e] += fmaxf(0.f, bnd - ps);  // hinge
      }
    }
  }

  // in-lane (8 elems) + cross-lane (wave32 butterfly) reduction
  float s = acc[0]+acc[1]+acc[2]+acc[3]+acc[4]+acc[5]+acc[6]+acc[7];
  #pragma unroll
  for (int off = 16; off > 0; off >>= 1)
    s += __shfl_xor(s, off, 32);

  if (lane == 0) wred[wave] = s;
  __syncthreads();
  if (tid == 0) {
    float bs = 0.f;
    #pragma unroll
    for (int w = 0; w < THREADS / 32; ++w) bs += wred[w];
    partials[blockIdx.x] = bs;   // deterministic: no float atomics
  }
}

// Fixed-order finalize: full-matrix sum == 2 * sum_{i<j}; normalize by
// 2*K = B*(B-1). Double accumulation over the 64 block partials.
__global__ void abrl_finalize(const float* __restrict__ partials,
                              float* __restrict__ out) {
  if (threadIdx.x == 0 && blockIdx.x == 0) {
    double s = 0.0;
    for (int i = 0; i < NBLOCKS; ++i) s += (double)partials[i];
    out[0] = (float)(s / ((double)BN * (double)(BN - 1)));
  }
}

extern "C" void kernel_launch(void* const* d_in, const int* in_sizes, int n_in,
                              void* d_out, int out_size, void* d_ws, size_t ws_size,
                              hipStream_t stream) {
  const float* pred   = (const float*)d_in[0];
  const float* target = (const float*)d_in[1];
  float* out      = (float*)d_out;
  float* partials = (float*)d_ws;   // needs NBLOCKS*4 = 256 bytes

  abrl_pairs<<<NBLOCKS, THREADS, 0, stream>>>(pred, target, partials);
  abrl_finalize<<<1, 32, 0, stream>>>(partials, out);
}